// LorenzModel_77867757076851
// MI455X (gfx1250) — compile-verified
//
#include <hip/hip_runtime.h>
#include <hip/hip_bf16.h>

// Lorenz explicit-Euler trajectory, 1e6 steps.
// Inherently serial nonlinear recurrence -> single wave32, single active lane,
// minimal dependent-op depth per step, bit-exact FP ordering vs the JAX
// reference (no FMA contraction), NT 128-bit row stores off the critical path.

typedef __attribute__((ext_vector_type(4))) float v4f;

__global__ void __launch_bounds__(32, 1)
lorenz_serial_kernel(const float* __restrict__ sigma,
                     const float* __restrict__ rho,
                     const float* __restrict__ beta,
                     const float* __restrict__ stats,
                     float* __restrict__ out,
                     int n) {
    // One lane owns the whole serial chain.
    if (threadIdx.x != 0 || blockIdx.x != 0) return;

    const float dt = 0.01f;                 // jnp.float32(0.01)
    const float s = sigma[0];
    const float r = rho[0];
    const float b = beta[0];

    float x = stats[0];
    float y = stats[1];
    float z = stats[2];

    v4f* __restrict__ orow = (v4f*)out;

    // Row 0 = the full stats parameter (including its 4th slot).
    {
        v4f row0;
        row0.x = x; row0.y = y; row0.z = z; row0.w = stats[3];
        __builtin_nontemporal_store(row0, orow);
    }

    // Rows 1..n-1: state after i Euler steps, time column = dt * i.
    // Exact reference op order (mul/add, round-to-nearest, NO fma contraction):
    //   nx = x + (s*(y-x))*dt
    //   ny = y + ((x*(r-z)) - y)*dt
    //   nz = z + ((x*y) - (b*z))*dt
#pragma unroll 4
    for (int i = 1; i < n; ++i) {
        const float nx = __fadd_rn(x,
            __fmul_rn(__fmul_rn(s, __fsub_rn(y, x)), dt));
        const float ny = __fadd_rn(y,
            __fmul_rn(__fsub_rn(__fmul_rn(x, __fsub_rn(r, z)), y), dt));
        const float nz = __fadd_rn(z,
            __fmul_rn(__fsub_rn(__fmul_rn(x, y), __fmul_rn(b, z)), dt));

        // Independent of the recurrence chain: co-issues with it.
        const float ti = __fmul_rn(dt, (float)i);

        x = nx; y = ny; z = nz;

        v4f row;
        row.x = x; row.y = y; row.z = z; row.w = ti;
        // 16B row, 16B aligned -> global_store_b128 with non-temporal hint;
        // 16 MB write-once stream, keep it out of near caches.
        __builtin_nontemporal_store(row, orow + i);
    }
}

extern "C" void kernel_launch(void* const* d_in, const int* in_sizes, int n_in,
                              void* d_out, int out_size, void* d_ws, size_t ws_size,
                              hipStream_t stream) {
    // setup_inputs() order: t, sigma, rho, beta, stats
    const float* sigma = (const float*)d_in[1];
    const float* rho   = (const float*)d_in[2];
    const float* beta  = (const float*)d_in[3];
    const float* stats = (const float*)d_in[4];
    float* out = (float*)d_out;

    const int n = in_sizes[0];   // T = 1,000,000 time steps

    // Serial chain: one wave32 is the whole machine's useful work here.
    lorenz_serial_kernel<<<1, 32, 0, stream>>>(sigma, rho, beta, stats, out, n);
}